// MarginalCalibrationPlot_5583457484864
// MI455X (gfx1250) — compile-verified
//
#include <hip/hip_runtime.h>
#include <hip/hip_bf16.h>
#include <stdint.h>

#define NBINS 15
#define THRESH 0.01f

typedef float vfloat4 __attribute__((ext_vector_type(4)));

// ---------------------------------------------------------------------------
// Fast path: specialized for C == 100 (reference shape).
// ---------------------------------------------------------------------------
namespace fastpath {

constexpr int       C       = 100;
constexpr int       CB      = C * NBINS;   // 1500 segments
constexpr int       THREADS = 256;
constexpr long long BLOCKS  = 512;         // compile-time: slab division is a magic-mul
constexpr int       LCAP    = 1280;        // staged labels per block (rows/block ~980)

// LDS histogram layout matches d_out layout:
//   [0, CB)      sum_p    -> becomes confs
//   [CB, 2CB)    sum_a    -> becomes accs
//   [2CB, 3CB)   count    -> n_samples (already final)

// Unconditional weighted update: invalid lanes add 0.0 (harmless), EXEC stays
// full -> no per-element exec-mask save/restore.
__device__ __forceinline__ void proc(float p, int col, float* h) {
  int bin = (int)(p * 15.0f);                    // floor for p >= 0
  bin = bin < (NBINS - 1) ? bin : (NBINS - 1);
  float left = (float)bin * (1.0f / 15.0f);
  // reference: p > THRESHOLD, 0 <= lo < B, p strictly > left bin edge
  bool valid = (p > THRESH) & (p < 1.0f) & (p > left);
  float w = valid ? 1.0f : 0.0f;
  int seg = col * NBINS + bin;
  atomicAdd(&h[seg], p * w);                     // ds_add_f32: sum_p
  atomicAdd(&h[2 * CB + seg], w);                // ds_add_f32: count
}

__global__ __launch_bounds__(THREADS)
void hist(const float* __restrict__ probas, const int* __restrict__ labels,
          float* __restrict__ out, long long tot4) {
  __shared__ float s_hist[3 * CB];
  __shared__ int   s_lab[LCAP];
  const int tid = threadIdx.x;

  for (int i = tid; i < 3 * CB; i += THREADS) s_hist[i] = 0.0f;

  // Contiguous slab per block -> coalesced b128 loads, contiguous label range.
  long long per = (tot4 + BLOCKS - 1) / BLOCKS;      // constant divisor
  long long b4  = (long long)blockIdx.x * per;
  long long e4  = b4 + per;
  if (e4 > tot4) e4 = tot4;

  int  row_first = 0;
  bool lds_lab   = false;
  if (b4 < e4) {
    row_first    = (int)((b4 * 4) / C);
    int row_last = (int)((e4 * 4 - 1) / C);
    int nrows    = row_last - row_first + 1;
    lds_lab      = (nrows <= LCAP);
    if (lds_lab) {
      // gfx1250 async global->LDS copy of this block's label slab.
      for (int i = tid; i < nrows; i += THREADS) {
        unsigned long long ga = (unsigned long long)(uintptr_t)(labels + row_first + i);
        unsigned la = (unsigned)(uintptr_t)(&s_lab[i]);  // low 32 bits = LDS offset
        asm volatile("global_load_async_to_lds_b32 %0, %1, off"
                     :: "v"(la), "v"(ga) : "memory");
      }
      asm volatile("s_wait_asynccnt 0" ::: "memory");
    }
  }
  __syncthreads();

  if (b4 < e4) {
    long long i4 = b4 + tid;
    if (i4 < e4) {
      long long e0 = i4 * 4;
      int row = (int)(e0 / C);
      int col = (int)(e0 - (long long)row * C);   // multiple of 4; f4 stays in-row
      const vfloat4* p4 = (const vfloat4*)probas;
      for (; i4 < e4; i4 += THREADS) {
        vfloat4 pv = __builtin_nontemporal_load(&p4[i4]);  // streaming, NT hint
        int lab = lds_lab ? s_lab[row - row_first] : labels[row];

        proc(pv.x, col + 0, s_hist);
        proc(pv.y, col + 1, s_hist);
        proc(pv.z, col + 2, s_hist);
        proc(pv.w, col + 3, s_hist);

        // sum_a: label matches at most one of the 4 consecutive columns.
        // Single rarely-taken branch per float4 (~4/100 threads).
        int dk = lab - col;
        if ((unsigned)dk < 4u) {
          float p = (dk == 0) ? pv.x : (dk == 1) ? pv.y : (dk == 2) ? pv.z : pv.w;
          int bin = (int)(p * 15.0f);
          bin = bin < (NBINS - 1) ? bin : (NBINS - 1);
          float left = (float)bin * (1.0f / 15.0f);
          if ((p > THRESH) & (p < 1.0f) & (p > left))
            atomicAdd(&s_hist[CB + lab * NBINS + bin], 1.0f);  // ds_add_f32: sum_a
        }

        col += (THREADS * 4) % C;                  // +24
        row += (THREADS * 4) / C;                  // +10
        if (col >= C) { col -= C; ++row; }
      }
    }
  }
  __syncthreads();

  // Flush per-block partials with relaxed agent-scope float atomics.
  for (int i = tid; i < 3 * CB; i += THREADS) {
    float v = s_hist[i];
    if (v != 0.0f)
      __hip_atomic_fetch_add(&out[i], v, __ATOMIC_RELAXED, __HIP_MEMORY_SCOPE_AGENT);
  }
}

} // namespace fastpath

// ---------------------------------------------------------------------------
// Generic fallback (any C): scalar grid-stride, global atomics.
// ---------------------------------------------------------------------------
__global__ void hist_generic(const float* __restrict__ probas,
                             const int* __restrict__ labels,
                             float* __restrict__ out, long long tot, int C) {
  long long i      = (long long)blockIdx.x * blockDim.x + threadIdx.x;
  long long stride = (long long)gridDim.x * blockDim.x;
  int cb = C * NBINS;
  for (; i < tot; i += stride) {
    float p = probas[i];
    int row = (int)(i / C);
    int col = (int)(i - (long long)row * C);
    int bin = (int)(p * 15.0f);
    bin = bin < (NBINS - 1) ? bin : (NBINS - 1);
    if (bin < 0) bin = 0;
    float left = (float)bin * (1.0f / 15.0f);
    if ((p > THRESH) & (p < 1.0f) & (p > left)) {
      int seg = col * NBINS + bin;
      __hip_atomic_fetch_add(&out[seg], p, __ATOMIC_RELAXED, __HIP_MEMORY_SCOPE_AGENT);
      __hip_atomic_fetch_add(&out[2 * cb + seg], 1.0f, __ATOMIC_RELAXED, __HIP_MEMORY_SCOPE_AGENT);
      if (labels[row] == col)
        __hip_atomic_fetch_add(&out[cb + seg], 1.0f, __ATOMIC_RELAXED, __HIP_MEMORY_SCOPE_AGENT);
    }
  }
}

__global__ void zero_out(float* __restrict__ out, int n) {
  int i = blockIdx.x * blockDim.x + threadIdx.x;
  if (i < n) out[i] = 0.0f;
}

__global__ void finalize(float* __restrict__ out, int cb) {
  int i = blockIdx.x * blockDim.x + threadIdx.x;
  if (i >= cb) return;
  float cnt = out[2 * cb + i];            // count == n_samples (final)
  if (cnt > 0.0f) {
    out[i]      = out[i] / cnt;           // confs = sum_p / count
    out[cb + i] = out[cb + i] / cnt;      // accs  = sum_a / count
  } else {
    float qn = __builtin_nanf("");
    out[i]      = qn;
    out[cb + i] = qn;
  }
}

extern "C" void kernel_launch(void* const* d_in, const int* in_sizes, int n_in,
                              void* d_out, int out_size, void* d_ws, size_t ws_size,
                              hipStream_t stream) {
  (void)n_in; (void)out_size; (void)d_ws; (void)ws_size;
  const float* probas = (const float*)d_in[0];
  const int*   labels = (const int*)d_in[1];
  float*       out    = (float*)d_out;

  long long tot = (long long)in_sizes[0];     // N * C
  long long n   = (long long)in_sizes[1];     // N
  int C  = (int)(tot / n);
  int cb = C * NBINS;
  int zn = 3 * cb;

  zero_out<<<(zn + 255) / 256, 256, 0, stream>>>(out, zn);

  if (C == 100 && (tot % 4) == 0) {
    fastpath::hist<<<(int)fastpath::BLOCKS, fastpath::THREADS, 0, stream>>>(
        probas, labels, out, tot / 4);
  } else {
    hist_generic<<<2048, 256, 0, stream>>>(probas, labels, out, tot, C);
  }

  finalize<<<(cb + 255) / 256, 256, 0, stream>>>(out, cb);
}